// MotionAggregatorGCN_67860483277502
// MI455X (gfx1250) — compile-verified
//
#include <hip/hip_runtime.h>
#include <hip/hip_bf16.h>

typedef __attribute__((ext_vector_type(2))) float v2f;
typedef __attribute__((ext_vector_type(8))) float v8f;

// ---------------------------------------------------------------------------
// fp32 GEMM with V_WMMA_F32_16X16X4_F32, templated on N,K so the unrolled
// fast path folds B addresses into global_load immediate offsets.
// C[M,N] = act(A[M,K] @ B[K,N] + bias[N]) ; row-major.
// Block = 128 threads = 4 waves; each wave owns one 16x16 tile; block = 64 cols.
// ---------------------------------------------------------------------------
#define GEMM_BK 64

template <int ACT, int NN, int KK>
__global__ __launch_bounds__(128) void gemm_wmma(const float* __restrict__ A,
                                                 const float* __restrict__ Bw,
                                                 const float* __restrict__ bias,
                                                 float* __restrict__ C, int M) {
  __shared__ float As[16][GEMM_BK + 4];   // row stride 68 floats
  const int tid   = threadIdx.x;
  const int lane  = tid & 31;
  const int wave  = tid >> 5;
  const int m0    = blockIdx.y * 16;
  const int n0    = (blockIdx.x * 4 + wave) * 16;
  const int khalf = lane >> 4;            // 0: K-pair {0,1}; 1: K-pair {2,3}
  const int ml    = lane & 15;
  const int nl    = lane & 15;
  const int nn    = n0 + nl;
  const bool nok  = nn < NN;

  v8f acc = {};

  for (int k0 = 0; k0 < KK; k0 += GEMM_BK) {
    __syncthreads();
    // Cooperative A-tile load, zero-padded in M and K tails.
    for (int i = tid; i < 16 * (GEMM_BK / 4); i += 128) {
      int r  = i / (GEMM_BK / 4);
      int c4 = (i % (GEMM_BK / 4)) * 4;
      float4 v = make_float4(0.f, 0.f, 0.f, 0.f);
      int gm = m0 + r, gk = k0 + c4;
      if (gm < M) {
        const float* ap = A + (size_t)gm * KK + gk;
        if (gk + 3 < KK) {
          v = *(const float4*)ap;
        } else {
          if (gk + 0 < KK) v.x = ap[0];
          if (gk + 1 < KK) v.y = ap[1];
          if (gk + 2 < KK) v.z = ap[2];
        }
      }
      *(float4*)&As[r][c4] = v;
    }
    __syncthreads();

    if (n0 >= NN) continue;                 // wave-uniform; barrier at loop top

    if (n0 + 16 <= NN && k0 + GEMM_BK <= KK) {
      // ---- fast path: no guards, constant B offsets, 16 WMMAs unrolled ----
      const float* Bp = Bw + ((size_t)k0 + 2 * khalf) * NN + nn;
      if (k0 + GEMM_BK < KK)
        __builtin_prefetch(Bp + (size_t)GEMM_BK * NN, 0, 1);
#pragma unroll
      for (int kk = 0; kk < GEMM_BK; kk += 4) {
        v2f a = *(const v2f*)&As[ml][kk + 2 * khalf];
        v2f b;
        b.x = __builtin_nontemporal_load(Bp + (size_t)kk * NN);
        b.y = __builtin_nontemporal_load(Bp + (size_t)(kk + 1) * NN);
        acc = __builtin_amdgcn_wmma_f32_16x16x4_f32(false, a, false, b,
                                                    (short)0, acc, false, false);
      }
    } else {
      // ---- tail path: N-tail tile and/or K-tail chunk, per-lane guards ----
      const int kc = (KK - k0 < GEMM_BK) ? (KK - k0) : GEMM_BK;
      for (int kk = 0; kk < kc; kk += 4) {
        const int kb = k0 + kk + 2 * khalf;
        v2f a = *(const v2f*)&As[ml][kk + 2 * khalf];
        v2f b;
        b.x = (nok && kb     < KK) ? __builtin_nontemporal_load(&Bw[(size_t)kb * NN + nn])       : 0.f;
        b.y = (nok && kb + 1 < KK) ? __builtin_nontemporal_load(&Bw[(size_t)(kb + 1) * NN + nn]) : 0.f;
        acc = __builtin_amdgcn_wmma_f32_16x16x4_f32(false, a, false, b,
                                                    (short)0, acc, false, false);
      }
    }
  }

  if (n0 < NN && nok) {
    const float bv = bias ? bias[nn] : 0.f;
#pragma unroll
    for (int r = 0; r < 8; ++r) {
      int mm = m0 + r + 8 * khalf;          // D layout: VGPR r -> M = r + 8*half
      if (mm < M) {
        float v = acc[r] + bv;
        if (ACT) v = fmaxf(v, 0.f);
        C[(size_t)mm * NN + nn] = v;
      }
    }
  }
}

// ---------------------------------------------------------------------------
// Implicit-im2col conv (256->128, 3x3, pad 1) as WMMA GEMM:
// out[128, 8448] = relu( Wc[128,2304] @ im2col[2304, 48*176] + bias[128] )
// B-tile staged in LDS (K-transposed) with an incremental (c,dy,dx) walker,
// so the wave inner loop is ds_load_b64 + ds_load_b64 + v_wmma.
// ---------------------------------------------------------------------------
__global__ __launch_bounds__(128) void conv_gemm_wmma(const float* __restrict__ Wc,
                                                      const float* __restrict__ img,
                                                      const float* __restrict__ bias,
                                                      float* __restrict__ out) {
  const int K = 2304, N = 8448, H = 48, W = 176, BK = 64;
  __shared__ float As[16][68];
  __shared__ float Bs[64][66];              // [col within block][krel]
  const int tid   = threadIdx.x;
  const int lane  = tid & 31;
  const int wave  = tid >> 5;
  const int m0    = blockIdx.y * 16;
  const int n0b   = blockIdx.x * 64;
  const int khalf = lane >> 4;
  const int ml    = lane & 15;
  const int wcol  = wave * 16 + (lane & 15);
  // staging coords: thread owns column sn for all chunks; krel = sk0 + 2*j
  const int sn  = tid & 63;
  const int sk0 = tid >> 6;                 // 0 or 1
  const int gn  = n0b + sn;
  const int sy  = gn / W, sx = gn % W;

  v8f acc = {};
  for (int k0 = 0; k0 < K; k0 += BK) {
    __syncthreads();
    // stage A tile (16 x 64), K=2304 and M=128 are exact multiples
    for (int i = tid; i < 256; i += 128) {
      int r = i >> 4, c4 = (i & 15) * 4;
      *(float4*)&As[r][c4] = *(const float4*)&Wc[(size_t)(m0 + r) * K + k0 + c4];
    }
    // stage B tile (64 x 64) via incremental im2col decode (one /9 per chunk)
    {
      int kb = k0 + sk0;
      int c = kb / 9, t = kb % 9;
      int dy = t / 3, dx = t - 3 * dy;
#pragma unroll 8
      for (int j = 0; j < 32; ++j) {
        int iy = sy + dy - 1, ix = sx + dx - 1;
        float v = (iy >= 0 && iy < H && ix >= 0 && ix < W)
                      ? img[((size_t)c * H + iy) * W + ix] : 0.f;
        Bs[sn][sk0 + 2 * j] = v;
        dx += 2;
        if (dx >= 3) { dx -= 3; ++dy; }
        if (dy >= 3) { dy -= 3; ++c; }
      }
    }
    __syncthreads();
#pragma unroll
    for (int kk = 0; kk < BK; kk += 4) {
      v2f a = *(const v2f*)&As[ml][kk + 2 * khalf];
      v2f b = *(const v2f*)&Bs[wcol][kk + 2 * khalf];
      acc = __builtin_amdgcn_wmma_f32_16x16x4_f32(false, a, false, b,
                                                  (short)0, acc, false, false);
    }
  }
  const int nn = n0b + wcol;
#pragma unroll
  for (int r = 0; r < 8; ++r) {
    int mm = m0 + r + 8 * khalf;            // mm = output channel
    out[(size_t)mm * N + nn] = fmaxf(acc[r] + bias[mm], 0.f);
  }
}

// --- small helper kernels ---------------------------------------------------

__global__ void prep_x(const float* __restrict__ f, const float* __restrict__ p,
                       float* __restrict__ xo, int n) {
  int i = blockIdx.x * blockDim.x + threadIdx.x;
  if (i < n) xo[i] = f[i] * p[i / 250];
}

__global__ void row_mean(const float* __restrict__ e, float* __restrict__ s,
                         int rows, int cols) {
  int j = blockIdx.x * blockDim.x + threadIdx.x;
  if (j < cols) {
    float a = 0.f;
    for (int r = 0; r < rows; ++r) a += e[(size_t)r * cols + j];
    s[j] = a / (float)rows;
  }
}

// chunked deterministic dot: part[chunk][128] = sum over 1024 k of conv*w
__global__ __launch_bounds__(128) void bev_fc_partial(const float* __restrict__ conv,
                                                      const float* __restrict__ w,
                                                      float* __restrict__ part) {
  __shared__ float cs[1024];
  const int n = threadIdx.x;
  const size_t k0 = (size_t)blockIdx.x * 1024;
  for (int i = n; i < 1024; i += 128) cs[i] = conv[k0 + i];
  __syncthreads();
  float acc = 0.f;
  const float* wp = w + k0 * 128 + n;
#pragma unroll 8
  for (int i = 0; i < 1024; ++i)
    acc += cs[i] * __builtin_nontemporal_load(&wp[(size_t)i * 128]);
  part[(size_t)blockIdx.x * 128 + n] = acc;
}

__global__ void bev_fc_reduce(const float* __restrict__ part,
                              const float* __restrict__ bias,
                              float* __restrict__ out, int chunks) {
  int n = threadIdx.x;
  float a = bias[n];
  for (int c = 0; c < chunks; ++c) a += part[(size_t)c * 128 + n];
  out[n] = a;
}

__global__ void map_prep(const float* __restrict__ mpf, const float* __restrict__ mask,
                         float* __restrict__ mf, int n) {
  int i = blockIdx.x * blockDim.x + threadIdx.x;
  if (i < n) {
    int o = i / 12544, p = (i % 12544) / 256;
    mf[i] = mpf[i] * mask[o * 49 + p];      // C=0 slice is the leading block
  }
}

__global__ void build_feat(const float* __restrict__ g, const float* __restrict__ bev,
                           const float* __restrict__ mape, float* __restrict__ feat,
                           int n) {
  int i = blockIdx.x * blockDim.x + threadIdx.x;
  if (i >= n) return;
  int o = i / 4224, j = i % 4224;
  float v;
  if (j < 3072)       v = g[j & 511];       // GCN output: identical per node
  else if (j < 3200)  v = bev[j - 3072];    // BEV embedding tiled over O
  else                v = mape[o * 1024 + (j - 3200)];
  feat[i] = v;
}

__global__ void softmax6(const float* __restrict__ logits, float* __restrict__ out) {
  int o = threadIdx.x;
  if (o < 16) {
    float m = -1e30f;
    for (int i = 0; i < 6; ++i) m = fmaxf(m, logits[o * 6 + i]);
    float e[6], s = 0.f;
    for (int i = 0; i < 6; ++i) { e[i] = __expf(logits[o * 6 + i] - m); s += e[i]; }
    for (int i = 0; i < 6; ++i) out[o * 6 + i] = e[i] / s;
  }
}

// ---------------------------------------------------------------------------

extern "C" void kernel_launch(void* const* d_in, const int* in_sizes, int n_in,
                              void* d_out, int out_size, void* d_ws, size_t ws_size,
                              hipStream_t stream) {
  const float* feats  = (const float*)d_in[0];
  const float* scores = (const float*)d_in[1];
  const float* bevimg = (const float*)d_in[2];
  const float* mpf    = (const float*)d_in[5];
  const float* mmask  = (const float*)d_in[6];
  const float* fe_w1 = (const float*)d_in[8],  *fe_b1 = (const float*)d_in[9];
  const float* fe_w2 = (const float*)d_in[10], *fe_b2 = (const float*)d_in[11];
  const float* fe_w3 = (const float*)d_in[12], *fe_b3 = (const float*)d_in[13];
  const float* gcn_w = (const float*)d_in[14], *gcn_b = (const float*)d_in[15];
  const float* conv_w = (const float*)d_in[16], *conv_b = (const float*)d_in[17];
  const float* bevfc_w = (const float*)d_in[18], *bevfc_b = (const float*)d_in[19];
  const float* me_w1 = (const float*)d_in[20], *me_b1 = (const float*)d_in[21];
  const float* me_w2 = (const float*)d_in[22], *me_b2 = (const float*)d_in[23];
  const float* me_w3 = (const float*)d_in[24], *me_b3 = (const float*)d_in[25];
  const float* td_w1 = (const float*)d_in[26], *td_b1 = (const float*)d_in[27];
  const float* td_w2 = (const float*)d_in[28], *td_b2 = (const float*)d_in[29];
  const float* td_w3 = (const float*)d_in[30], *td_b3 = (const float*)d_in[31];
  const float* ce_w1 = (const float*)d_in[32], *ce_b1 = (const float*)d_in[33];
  const float* ce_w2 = (const float*)d_in[34], *ce_b2 = (const float*)d_in[35];
  const float* ce_w3 = (const float*)d_in[36], *ce_b3 = (const float*)d_in[37];
  float* out = (float*)d_out;
  float* ws  = (float*)d_ws;

  // workspace layout (floats)
  const size_t o_x = 0;           // 96000  : weighted trajectories (384,250)
  const size_t o_h1 = 96000;      // 96000  : fe hidden (h2 reuses o_x)
  const size_t o_emb = 192000;    // 196608 : node embeddings (384,512)
  const size_t o_s = 388608;      // 512    : mean node feature
  const size_t o_g = 389120;      // 512    : GCN output vector
  const size_t o_conv = 389632;   // 1081344: conv output (128,8448)
  const size_t o_part = 1470976;  // 135168 : bev fc partials (1056,128)
  const size_t o_bev = 1606144;   // 128    : bev embedding
  const size_t o_mf = 1606272;    // 200704 : map features (reused as mh2)
  const size_t o_mh1 = 1806976;   // 200704
  const size_t o_mape = 2007680;  // 16384  : map embedding (16,1024)
  const size_t o_feat = 2024064;  // 67584  : concat feature (16,4224)
  const size_t o_t1 = 2091648;    // 67584
  const size_t o_t2 = 2159232;    // 67584
  const size_t o_logit = 2226816; // 96

  auto cdiv = [](int a, int b) { return (a + b - 1) / b; };
#define GEMM(ACT, NN, KK, A, B, BIAS, CC, M)                                   \
  hipLaunchKernelGGL((gemm_wmma<ACT, NN, KK>), dim3(cdiv(NN, 64), cdiv(M, 16)),\
                     dim3(128), 0, stream, A, B, BIAS, CC, M)

  // 1) feature encoder: (384,250) -> 250 -> 250 -> 512
  prep_x<<<cdiv(96000, 256), 256, 0, stream>>>(feats, scores, ws + o_x, 96000);
  GEMM(1, 250, 250, ws + o_x, fe_w1, fe_b1, ws + o_h1, 384);
  GEMM(1, 250, 250, ws + o_h1, fe_w2, fe_b2, ws + o_x, 384);
  GEMM(0, 512, 250, ws + o_x, fe_w3, fe_b3, ws + o_emb, 384);

  // 2) GCN on the complete graph collapses to mean + matvec (all nodes equal)
  row_mean<<<cdiv(512, 256), 256, 0, stream>>>(ws + o_emb, ws + o_s, 384, 512);
  GEMM(0, 512, 512, ws + o_s, gcn_w, gcn_b, ws + o_g, 1);

  // 3) BEV: conv as implicit-im2col WMMA GEMM, then chunked 1M-dot FC
  conv_gemm_wmma<<<dim3(132, 8), 128, 0, stream>>>(conv_w, bevimg, conv_b, ws + o_conv);
  bev_fc_partial<<<1056, 128, 0, stream>>>(ws + o_conv, bevfc_w, ws + o_part);
  bev_fc_reduce<<<1, 128, 0, stream>>>(ws + o_part, bevfc_b, ws + o_bev, 1056);

  // 4) map encoder: (16,12544) -> 12544 -> 12544 -> 1024
  map_prep<<<cdiv(200704, 256), 256, 0, stream>>>(mpf, mmask, ws + o_mf, 200704);
  GEMM(1, 12544, 12544, ws + o_mf, me_w1, me_b1, ws + o_mh1, 16);
  GEMM(1, 12544, 12544, ws + o_mh1, me_w2, me_b2, ws + o_mf, 16);
  GEMM(0, 1024, 12544, ws + o_mf, me_w3, me_b3, ws + o_mape, 16);

  // 5) concat + decoders
  build_feat<<<cdiv(67584, 256), 256, 0, stream>>>(ws + o_g, ws + o_bev,
                                                   ws + o_mape, ws + o_feat, 67584);
  GEMM(1, 4224, 4224, ws + o_feat, td_w1, td_b1, ws + o_t1, 16);
  GEMM(1, 4224, 4224, ws + o_t1, td_w2, td_b2, ws + o_t2, 16);
  GEMM(0, 1500, 4224, ws + o_t2, td_w3, td_b3, out, 16);       // traj

  GEMM(1, 4224, 4224, ws + o_feat, ce_w1, ce_b1, ws + o_t1, 16);
  GEMM(1, 4224, 4224, ws + o_t1, ce_w2, ce_b2, ws + o_t2, 16);
  GEMM(0, 6, 4224, ws + o_t2, ce_w3, ce_b3, ws + o_logit, 16);
  softmax6<<<1, 16, 0, stream>>>(ws + o_logit, out + 24000);   // conf
#undef GEMM
}